// TempoBaseHDO_65816078844463
// MI455X (gfx1250) — compile-verified
//
#include <hip/hip_runtime.h>
#include <math.h>

#define Tdim 8192
#define Bdim 32
#define Ddim 256
#define Kwin 16
#define NWIN 512  // Tdim / Kwin

typedef __attribute__((ext_vector_type(2))) float v2f;
typedef __attribute__((ext_vector_type(8))) float v8f;

// ---------------------------------------------------------------------------
// Pass 0: pre-pack W[256][256] into WMMA B-fragment order:
//   wpack[((nt*64 + kk)*32 + lane)] = (v2f){ W[4kk+2*khalf  ][nt*16+m16],
//                                            W[4kk+2*khalf+1][nt*16+m16] }
// so the GEMM hot loop does ONE coalesced global_load_b64 per v_wmma.
// 256 KB once; trivial vs. the 0.67 GB roofline.
// ---------------------------------------------------------------------------
__global__ void pass0_packw(const float* __restrict__ Wm,
                            float* __restrict__ wpack) {
  const int nt = blockIdx.x;            // 0..15 N-tile
  const int lane = threadIdx.x & 31;
  const int m16 = lane & 15;
  const int khalf = lane >> 4;
#pragma unroll
  for (int kk = (int)(threadIdx.x >> 5); kk < 64; kk += 8) {
    const int kbase = 4 * kk + 2 * khalf;
    v2f v;
    v[0] = Wm[(size_t)kbase * Ddim + nt * 16 + m16];
    v[1] = Wm[(size_t)(kbase + 1) * Ddim + nt * 16 + m16];
    ((v2f*)wpack)[((size_t)nt * 64 + kk) * 32 + lane] = v;
  }
}

// ---------------------------------------------------------------------------
// Pass 1: per (window w, sample b): sig[d] = mean_k x[wK+k][b][d]
//         vtemp = mean_d std_k(x, ddof=1)
// One block (256 threads, thread==d) per (w,b). x read once, coalesced on d.
// ---------------------------------------------------------------------------
__global__ void pass1_sig(const float* __restrict__ x,
                          float* __restrict__ sig,
                          float* __restrict__ vtemp) {
  const int wb = blockIdx.x;            // 0 .. NWIN*Bdim-1
  const int w = wb >> 5, b = wb & 31;
  const int d = threadIdx.x;            // 0 .. 255
  const float* base = x + ((size_t)((size_t)w * Kwin) * Bdim + b) * Ddim + d;
  float s1 = 0.f, s2 = 0.f;
#pragma unroll
  for (int k = 0; k < Kwin; ++k) {
    float v = base[(size_t)k * Bdim * Ddim];
    s1 += v;
    s2 += v * v;
  }
  const float mean = s1 * (1.0f / Kwin);
  sig[(size_t)wb * Ddim + d] = mean;
  float var = (s2 - s1 * mean) * (1.0f / (Kwin - 1));  // ddof = 1
  var = var < 0.f ? 0.f : var;
  float v = sqrtf(var);
  // block-reduce sum over 256 threads (wave32 shuffles + LDS)
  __shared__ float red[8];
#pragma unroll
  for (int off = 16; off > 0; off >>= 1) v += __shfl_xor(v, off);
  if ((threadIdx.x & 31) == 0) red[threadIdx.x >> 5] = v;
  __syncthreads();
  if (threadIdx.x == 0) {
    float t = 0.f;
#pragma unroll
    for (int i = 0; i < 8; ++i) t += red[i];
    vtemp[wb] = t * (1.0f / Ddim);
  }
}

// ---------------------------------------------------------------------------
// Pass 2: sequential gating scan over 512 windows. One block, 32 waves;
// wave i owns sample b=i, cached signature lives in registers (8 f32/lane).
// Emits refresh flag + drive source window per (w,b).
// ---------------------------------------------------------------------------
__global__ void pass2_scan(const float* __restrict__ sig,
                           const float* __restrict__ vtemp,
                           int* __restrict__ refresh_flag,
                           int* __restrict__ src_win) {
  const int b = threadIdx.x >> 5;
  const int lane = threadIdx.x & 31;
  float prev[8];
#pragma unroll
  for (int j = 0; j < 8; ++j) prev[j] = 0.f;
  int valid = 0, age = 0, src = 0;
  for (int w = 0; w < NWIN; ++w) {
    const float* sc = sig + ((size_t)w * Bdim + b) * Ddim;
    float cur[8];
    float dsum = 0.f, psum = 0.f;
#pragma unroll
    for (int j = 0; j < 8; ++j) {
      cur[j] = sc[j * 32 + lane];
      float df = cur[j] - prev[j];
      dsum += df * df;
      psum += prev[j] * prev[j];
    }
#pragma unroll
    for (int off = 16; off > 0; off >>= 1) {
      dsum += __shfl_xor(dsum, off);
      psum += __shfl_xor(psum, off);
    }
    const float d_inter = sqrtf(dsum) / (sqrtf(psum) + 1e-6f);
    const float vt = vtemp[w * Bdim + b];
    const bool hard = d_inter > 1.46f;                 // TAU_INVALID
    const bool veff = (valid != 0) && !hard;
    const int age_eff = hard ? 0 : age;
    const bool reuse = veff && (d_inter <= 1.42f)      // TAU_REUSE
                            && (vt <= 1.05f)           // VTEMP_MAX
                            && (age_eff < 4);          // MAX_AGE
    if (!reuse) {  // refresh
#pragma unroll
      for (int j = 0; j < 8; ++j) prev[j] = cur[j];
      src = w; valid = 1; age = 0;
    } else {
      valid = 1; age = age_eff + 1;
    }
    if (lane == 0) {
      refresh_flag[w * Bdim + b] = reuse ? 0 : 1;
      src_win[w * Bdim + b] = reuse ? src : w;
    }
  }
}

// ---------------------------------------------------------------------------
// Pass 3a: fresh path. One wave per (w,b) tile that needs refresh:
//   Y[16x256] = relu(A[16x256] * W[256x256] + bias), drive = mean over 16 rows.
// A held fully in registers (64 x v2f). 1024 v_wmma_f32_16x16x4_f32 per tile;
// B fragments are single coalesced b64 loads from the packed W.
// Wave-uniform early-out keeps EXEC all-ones around WMMA.
// ---------------------------------------------------------------------------
__global__ void __launch_bounds__(128) pass3a_gemm(
    const float* __restrict__ x, const float* __restrict__ wpack,
    const float* __restrict__ bias, const int* __restrict__ refresh_flag,
    float* __restrict__ out, float* __restrict__ drive) {
  const int wave = blockIdx.x * 4 + (threadIdx.x >> 5);
  const int w = wave >> 5;   // window  (b fastest -> waves in a block share w)
  const int b = wave & 31;   // sample
  if (!refresh_flag[w * Bdim + b]) return;  // reuse tile: pass3b handles it

  const int lane = threadIdx.x & 31;
  const int m16 = lane & 15;     // row within half / column within tile
  const int khalf = lane >> 4;   // K/M half selector

  // Load A: row m16 of the window, all 256 K values this lane needs.
  // Fragment kk covers K columns [4kk, 4kk+3]; lane half picks {0,1} or {2,3}.
  const float* xrow =
      x + ((size_t)((size_t)w * Kwin + m16) * Bdim + b) * Ddim;
  v2f afrag[64];
#pragma unroll
  for (int kk = 0; kk < 64; ++kk)
    afrag[kk] = *(const v2f*)(xrow + 4 * kk + 2 * khalf);

  const v2f* wp = (const v2f*)wpack;
#pragma unroll 1
  for (int nt = 0; nt < 16; ++nt) {
    const v2f* wnt = wp + ((size_t)nt * 64) * 32 + lane;
    v8f c = {};
#pragma unroll
    for (int kk = 0; kk < 64; ++kk) {
      const v2f bf = wnt[kk * 32];  // one coalesced b64 per wmma
      // D = A(16x4) * B(4x16) + C ; 8 args: neg_a, A, neg_b, B, c_mod, C, ra, rb
      c = __builtin_amdgcn_wmma_f32_16x16x4_f32(false, afrag[kk], false, bf,
                                                (short)0, c, false, false);
    }
    // Epilogue: bias + relu, store 16x16 tile, accumulate drive (row mean).
    const int col = nt * 16 + m16;
    const float bv = bias[col];
    float dsum = 0.f;
#pragma unroll
    for (int r = 0; r < 8; ++r) {
      float y = c[r] + bv;
      y = y > 0.f ? y : 0.f;
      const int m = r + 8 * khalf;  // C layout: vgpr r -> rows r / r+8
      out[((size_t)((size_t)w * Kwin + m) * Bdim + b) * Ddim + col] = y;
      dsum += y;
    }
    dsum += __shfl_xor(dsum, 16);   // combine rows 0-7 with rows 8-15
    if (khalf == 0)
      drive[((size_t)w * Bdim + b) * Ddim + col] = dsum * (1.0f / Kwin);
  }
}

// ---------------------------------------------------------------------------
// Pass 3b: reuse path. Broadcast cached drive of the source window over the
// 16 output rows. Separate launch => source drives are already written.
// ---------------------------------------------------------------------------
__global__ void pass3b_bcast(const int* __restrict__ refresh_flag,
                             const int* __restrict__ src_win,
                             const float* __restrict__ drive,
                             float* __restrict__ out) {
  const int wb = blockIdx.x;
  if (refresh_flag[wb]) return;   // fresh tile: already written by pass3a
  const int w = wb >> 5, b = wb & 31;
  const int e = threadIdx.x;
  const int src = src_win[wb];
  const float val = drive[((size_t)src * Bdim + b) * Ddim + e];
#pragma unroll
  for (int k = 0; k < Kwin; ++k)
    out[((size_t)((size_t)w * Kwin + k) * Bdim + b) * Ddim + e] = val;
}

// ---------------------------------------------------------------------------
extern "C" void kernel_launch(void* const* d_in, const int* in_sizes, int n_in,
                              void* d_out, int out_size, void* d_ws,
                              size_t ws_size, hipStream_t stream) {
  const float* x = (const float*)d_in[0];    // [8192,32,256]
  const float* W = (const float*)d_in[1];    // [256,256]
  const float* bias = (const float*)d_in[2]; // [256]
  float* out = (float*)d_out;                // [8192,32,256]

  char* ws = (char*)d_ws;
  float* sig = (float*)ws;    ws += (size_t)NWIN * Bdim * Ddim * sizeof(float); // 16 MB
  float* drive = (float*)ws;  ws += (size_t)NWIN * Bdim * Ddim * sizeof(float); // 16 MB
  float* wpack = (float*)ws;  ws += (size_t)Ddim * Ddim * sizeof(float);        // 256 KB
  float* vtemp = (float*)ws;  ws += (size_t)NWIN * Bdim * sizeof(float);
  int* flag = (int*)ws;       ws += (size_t)NWIN * Bdim * sizeof(int);
  int* src = (int*)ws;

  pass0_packw<<<16, 256, 0, stream>>>(W, wpack);
  pass1_sig<<<NWIN * Bdim, 256, 0, stream>>>(x, sig, vtemp);
  pass2_scan<<<1, 1024, 0, stream>>>(sig, vtemp, flag, src);
  pass3a_gemm<<<NWIN * Bdim / 4, 128, 0, stream>>>(x, wpack, bias, flag, out,
                                                   drive);
  pass3b_bcast<<<NWIN * Bdim, 256, 0, stream>>>(flag, src, drive, out);
}